// SymmetricContraction_59236188946491
// MI455X (gfx1250) — compile-verified
//
#include <hip/hip_runtime.h>

typedef __attribute__((ext_vector_type(2))) float v2f;
typedef __attribute__((ext_vector_type(8))) float v8f;

#define NN   4096
#define CC   128
#define II   9
#define DD   9
#define EE   10
#define PP3  23
#define PP2  5
#define PP1  3

#define NP      736      // dij (=729) padded to 46*16
#define KP      12       // k (=9) padded to 3 WMMA k-steps of 4
#define NTILES  46       // NP/16
#define PADN    4112     // per-species node-list stride (N + 16)
#define MAXTILES 272     // >= ceil(N/16) + E
#define ASTRIDE 737      // LDS row stride (737 mod 64 = 33 -> conflict-free)

// ---------------------------------------------------------------------------
// Kernel 1: deterministic grouping of nodes by species + tile table.
// ---------------------------------------------------------------------------
__global__ void build_lists(const int* __restrict__ idx, int* __restrict__ nodeList,
                            int* __restrict__ counts, int* __restrict__ tileSpec,
                            int* __restrict__ tileBase, int* __restrict__ tileValid,
                            int* __restrict__ nTiles) {
    int e = threadIdx.x;
    if (e < EE) {
        int cnt = 0;
        for (int b = 0; b < NN; ++b)
            if (idx[b] == e) nodeList[e * PADN + cnt++] = b;
        counts[e] = cnt;
        int up = (cnt + 15) & ~15;
        int last = (cnt > 0) ? nodeList[e * PADN + cnt - 1] : 0;
        for (int j = cnt; j < up; ++j) nodeList[e * PADN + j] = last;
    }
    __syncthreads();
    if (threadIdx.x == 0) {
        int T = 0;
        for (int e2 = 0; e2 < EE; ++e2) {
            int cnt = counts[e2];
            int tc  = (cnt + 15) >> 4;
            for (int t = 0; t < tc; ++t) {
                tileSpec[T]  = e2;
                tileBase[T]  = e2 * PADN + t * 16;
                int v = cnt - t * 16; if (v > 16) v = 16;
                tileValid[T] = v;
                ++T;
            }
        }
        nTiles[0] = T;
    }
}

// ---------------------------------------------------------------------------
// Kernel 2: U3c[e][c][k][n] = sum_p u3[d,i,j,k,p] * w3[e,p,c]  (zero-padded)
// Layout: [E][C][KP][NP], k-major so a B-tile row is contiguous.
// ---------------------------------------------------------------------------
__global__ void precomp_u3(const float* __restrict__ u3, const float* __restrict__ w3,
                           float* __restrict__ U3c) {
    size_t gid = (size_t)blockIdx.x * blockDim.x + threadIdx.x;
    const size_t total = (size_t)EE * CC * KP * NP;
    if (gid >= total) return;
    int n = (int)(gid % NP);
    int k = (int)((gid / NP) % KP);
    int c = (int)((gid / ((size_t)NP * KP)) % CC);
    int e = (int)(gid / ((size_t)NP * KP * CC));
    float acc = 0.f;
    if (n < DD * II * II && k < II) {
        int d = n / (II * II);
        int i = (n / II) % II;
        int j = n % II;
        const float* up = u3 + ((((size_t)d * II + i) * II + j) * II + k) * PP3;
        const float* wp = w3 + ((size_t)e * PP3) * CC + c;
#pragma unroll
        for (int p = 0; p < PP3; ++p) acc += up[p] * wp[(size_t)p * CC];
    }
    U3c[gid] = acc;
}

// Kernel 3: U2c[e][c][n] = sum_p u2[d,i,j,p] * w2[e,p,c]  (n padded to NP)
__global__ void precomp_u2(const float* __restrict__ u2, const float* __restrict__ w2,
                           float* __restrict__ U2c) {
    size_t gid = (size_t)blockIdx.x * blockDim.x + threadIdx.x;
    const size_t total = (size_t)EE * CC * NP;
    if (gid >= total) return;
    int n = (int)(gid % NP);
    int c = (int)((gid / NP) % CC);
    int e = (int)(gid / ((size_t)NP * CC));
    float acc = 0.f;
    if (n < DD * II * II) {
        const float* up = u2 + (size_t)n * PP2;      // u2 is [D,I,I,P2], n == d*81+i*9+j
        const float* wp = w2 + ((size_t)e * PP2) * CC + c;
#pragma unroll
        for (int p = 0; p < PP2; ++p) acc += up[p] * wp[(size_t)p * CC];
    }
    U2c[gid] = acc;
}

// Kernel 4: U1c[e][c][di] = sum_p u1[d,i,p] * w1[e,p,c]  (di padded to 96)
__global__ void precomp_u1(const float* __restrict__ u1, const float* __restrict__ w1,
                           float* __restrict__ U1c) {
    size_t gid = (size_t)blockIdx.x * blockDim.x + threadIdx.x;
    const size_t total = (size_t)EE * CC * 96;
    if (gid >= total) return;
    int n = (int)(gid % 96);
    int c = (int)((gid / 96) % CC);
    int e = (int)(gid / ((size_t)96 * CC));
    float acc = 0.f;
    if (n < DD * II) {
        const float* up = u1 + (size_t)n * PP1;      // u1 is [D,I,P1], n == d*9+i
        const float* wp = w1 + ((size_t)e * PP1) * CC + c;
#pragma unroll
        for (int p = 0; p < PP1; ++p) acc += up[p] * wp[(size_t)p * CC];
    }
    U1c[gid] = acc;
}

// ---------------------------------------------------------------------------
// Main kernel: per (node-tile-of-16, channel) block.
//   Stage 1 (WMMA): A[16m x 736n] = X[16m x 12k] * U3c[12k x 736n]
//   Stage 2 (VALU): out[m,c,d] = sum_i ( sum_j (A + U2)*x_j + U1 ) * x_i
// ---------------------------------------------------------------------------
__global__ __launch_bounds__(256)
void sym_contract_main(const float* __restrict__ x, const float* __restrict__ U3c,
                       const float* __restrict__ U2c, const float* __restrict__ U1c,
                       const int* __restrict__ nodeList, const int* __restrict__ tileSpec,
                       const int* __restrict__ tileBase, const int* __restrict__ tileValid,
                       const int* __restrict__ nTiles, float* __restrict__ out) {
    __shared__ float Atile[16 * ASTRIDE];
    __shared__ float xs[16 * KP];
    __shared__ int   nids[16];

    if ((int)blockIdx.x >= nTiles[0]) return;
    const int e     = tileSpec[blockIdx.x];
    const int base  = tileBase[blockIdx.x];
    const int valid = tileValid[blockIdx.x];
    const int c     = blockIdx.y;
    const int t     = threadIdx.x;

    if (t < 16) nids[t] = nodeList[base + t];
    __syncthreads();
    if (t < 16 * KP) {
        int m = t / KP, k = t % KP;
        xs[t] = (k < II) ? x[((size_t)nids[m] * CC + c) * II + k] : 0.f;
    }
    __syncthreads();

    const int wave = t >> 5, lane = t & 31;
    const int col  = lane & 15;          // M for A-frag, N for B/C-frags
    const int hi   = lane >> 4;          // lane half selects K (A/B) or M+8 (C)
    const float* __restrict__ Ub = U3c + ((size_t)e * CC + c) * ((size_t)KP * NP);

    for (int nt = wave; nt < NTILES; nt += 8) {        // uniform per wave: EXEC all-ones
        v8f acc = {0.f, 0.f, 0.f, 0.f, 0.f, 0.f, 0.f, 0.f};
#pragma unroll
        for (int s = 0; s < 3; ++s) {                  // K = 12 in steps of 4
            const int k0 = s * 4 + 2 * hi;             // ISA A-layout: k = vgpr + 2*hi
            v2f a, b;
            a[0] = xs[col * KP + k0];
            a[1] = xs[col * KP + k0 + 1];
            b[0] = Ub[(size_t)k0 * NP + nt * 16 + col];
            b[1] = Ub[(size_t)(k0 + 1) * NP + nt * 16 + col];
            acc = __builtin_amdgcn_wmma_f32_16x16x4_f32(
                false, a, false, b, (short)0, acc, false, false);
        }
#pragma unroll
        for (int r = 0; r < 8; ++r)                    // C-layout: M = r + 8*hi, N = col
            Atile[(r + 8 * hi) * ASTRIDE + nt * 16 + col] = acc[r];
    }
    __syncthreads();

    if (t < 16 * DD) {
        const int m = t & 15, d = t >> 4;
        const float* __restrict__ U2p = U2c + ((size_t)e * CC + c) * NP + d * (II * II);
        const float* __restrict__ U1p = U1c + ((size_t)e * CC + c) * 96 + d * II;
        const float* xm = &xs[m * KP];
        const float* Am = &Atile[m * ASTRIDE + d * (II * II)];
        float accd = 0.f;
#pragma unroll
        for (int i = 0; i < II; ++i) {
            float s2 = 0.f;
#pragma unroll
            for (int j = 0; j < II; ++j)
                s2 += (Am[i * II + j] + U2p[i * II + j]) * xm[j];
            accd += (s2 + U1p[i]) * xm[i];
        }
        if (m < valid)
            out[((size_t)nids[m] * CC + c) * DD + d] = accd;
    }
}

// ---------------------------------------------------------------------------
extern "C" void kernel_launch(void* const* d_in, const int* in_sizes, int n_in,
                              void* d_out, int out_size, void* d_ws, size_t ws_size,
                              hipStream_t stream) {
    const float* x   = (const float*)d_in[0];
    const int*   idx = (const int*)  d_in[1];
    const float* u3  = (const float*)d_in[2];
    const float* u2  = (const float*)d_in[3];
    const float* u1  = (const float*)d_in[4];
    const float* w3  = (const float*)d_in[5];
    const float* w2  = (const float*)d_in[6];
    const float* w1  = (const float*)d_in[7];
    float* out = (float*)d_out;

    // workspace carve-up (~49.5 MB)
    float* U3c = (float*)d_ws;                                  // E*C*KP*NP
    float* U2c = U3c + (size_t)EE * CC * KP * NP;               // E*C*NP
    float* U1c = U2c + (size_t)EE * CC * NP;                    // E*C*96
    int* nodeList  = (int*)(U1c + (size_t)EE * CC * 96);        // E*PADN
    int* counts    = nodeList + (size_t)EE * PADN;              // 16
    int* tileSpec  = counts + 16;                               // MAXTILES
    int* tileBase  = tileSpec + MAXTILES;                       // MAXTILES
    int* tileValid = tileBase + MAXTILES;                       // MAXTILES
    int* nTiles    = tileValid + MAXTILES;                      // 1

    build_lists<<<1, 32, 0, stream>>>(idx, nodeList, counts, tileSpec, tileBase,
                                      tileValid, nTiles);
    {
        size_t total = (size_t)EE * CC * KP * NP;
        precomp_u3<<<(unsigned)((total + 255) / 256), 256, 0, stream>>>(u3, w3, U3c);
    }
    {
        size_t total = (size_t)EE * CC * NP;
        precomp_u2<<<(unsigned)((total + 255) / 256), 256, 0, stream>>>(u2, w2, U2c);
    }
    {
        size_t total = (size_t)EE * CC * 96;
        precomp_u1<<<(unsigned)((total + 255) / 256), 256, 0, stream>>>(u1, w1, U1c);
    }
    dim3 grid(MAXTILES, CC);
    sym_contract_main<<<grid, 256, 0, stream>>>(x, U3c, U2c, U1c, nodeList, tileSpec,
                                                tileBase, tileValid, nTiles, out);
}